// DiscreteEmbedding_69552700392130
// MI455X (gfx1250) — compile-verified
//
#include <hip/hip_runtime.h>

#define WAVES_PER_BLOCK 8
#define THREADS (WAVES_PER_BLOCK * 32)
#define OUT_D 128

typedef float v4f __attribute__((ext_vector_type(4)));
typedef int   v4i __attribute__((ext_vector_type(4)));

#if defined(__gfx1250__) && __has_builtin(__builtin_amdgcn_global_load_async_to_lds_b128)
#define USE_ASYNC_LDS 1
#else
#define USE_ASYNC_LDS 0
#endif

// One wave per output row. Row needs embedding rows k and/or k+1 with weights
// w0 = 0.5*(1 - sign(d-0.5)), w1 = 0.5*(1 + sign(d-0.5)), d = xs - floor(xs),
// which reproduces the reference rect-window (sign(0)=0 -> 0.5/0.5 split).
// Memory-bound: 32MB store floor; embedding (1MB) is L2-resident.
__global__ __launch_bounds__(THREADS) void
discrete_embedding_gather(const float* __restrict__ x,
                          const float* __restrict__ emb,
                          float* __restrict__ out,
                          int nrows, int npoints)
{
#if USE_ASYNC_LDS
    __shared__ __align__(16) float lds[WAVES_PER_BLOCK][2][OUT_D]; // 8 KB/block
#endif
    const int lane   = threadIdx.x & 31;
    const int wave   = threadIdx.x >> 5;
    const int stride = gridDim.x * WAVES_PER_BLOCK;

    for (int row = blockIdx.x * WAVES_PER_BLOCK + wave; row < nrows; row += stride) {
        // Prefetch next iteration's x (global_prefetch_b8 on gfx1250).
        if (row + stride < nrows) __builtin_prefetch(&x[row + stride], 0, 1);

        float xs = x[row] * (float)npoints;     // exact: scale by 2^11
        float kf = floorf(xs);
        int   k  = (int)kf;
        if (k < 0) k = 0;
        if (k >= npoints) k = npoints - 1;      // paranoia; x in [0,1) keeps k in range
        float d  = xs - kf;                     // exact (Sterbenz)
        float s  = (d > 0.5f) ? 1.0f : ((d < 0.5f) ? -1.0f : 0.0f); // sign(d-0.5)
        float w0 = 0.5f * (1.0f - s);
        float w1 = (k + 1 < npoints) ? (0.5f * (1.0f + s)) : 0.0f;
        const bool need0 = (w0 != 0.0f);        // wave-uniform branches
        const bool need1 = (w1 != 0.0f);

        const float* g0 = emb + (size_t)k * OUT_D + lane * 4;
        const float* g1 = emb + (size_t)(k + 1) * OUT_D + lane * 4;

        v4f acc = {0.0f, 0.0f, 0.0f, 0.0f};
#if USE_ASYNC_LDS
        // Drain prior LDS reads before overwriting the staging slots: async-pipe
        // LDS writes are NOT ordered with this wave's DS-pipe reads (WAR hazard
        // across iterations). DScnt is normally already 0 here, so this is free.
        asm volatile("s_wait_dscnt 0x0" ::: "memory");

        // Stage needed row(s) via CDNA5 async global->LDS DMA path (ASYNCcnt).
        if (need0)
            __builtin_amdgcn_global_load_async_to_lds_b128(
                (v4i*)g0, (v4i*)(&lds[wave][0][lane * 4]), 0, 0);
        if (need1)
            __builtin_amdgcn_global_load_async_to_lds_b128(
                (v4i*)g1, (v4i*)(&lds[wave][1][lane * 4]), 0, 0);
#if __has_builtin(__builtin_amdgcn_s_wait_asynccnt)
        __builtin_amdgcn_s_wait_asynccnt(0);
#else
        asm volatile("s_wait_asynccnt 0" ::: "memory");
#endif
        if (need0) {
            v4f e0 = *(const v4f*)(&lds[wave][0][lane * 4]);
            acc = w0 * e0;
        }
        if (need1) {
            v4f e1 = *(const v4f*)(&lds[wave][1][lane * 4]);
            acc += w1 * e1;
        }
#else
        if (need0) {
            v4f e0 = *(const v4f*)g0;           // global_load_b128, coalesced 512B
            acc = w0 * e0;
        }
        if (need1) {
            v4f e1 = *(const v4f*)g1;
            acc += w1 * e1;
        }
#endif
        *(v4f*)(out + (size_t)row * OUT_D + lane * 4) = acc; // global_store_b128
    }
}

extern "C" void kernel_launch(void* const* d_in, const int* in_sizes, int n_in,
                              void* d_out, int out_size, void* d_ws, size_t ws_size,
                              hipStream_t stream) {
    const float* x   = (const float*)d_in[0];   // [B,S] f32, 65536 elems
    const float* emb = (const float*)d_in[1];   // [P,D] f32, 2048*128
    float* out       = (float*)d_out;           // [B,S,D] f32

    const int nrows   = in_sizes[0];            // 65536
    const int npoints = in_sizes[1] / OUT_D;    // 2048

    const int blocks = 2048;                    // 16384 waves, 4 rows each
    discrete_embedding_gather<<<blocks, THREADS, 0, stream>>>(x, emb, out, nrows, npoints);
}